// GCN_23330262352099
// MI455X (gfx1250) — compile-verified
//
#include <hip/hip_runtime.h>

typedef float v2f __attribute__((ext_vector_type(2)));
typedef float v8f __attribute__((ext_vector_type(8)));

// ---------------------------------------------------------------------------
// Degree / normalization kernels
// ---------------------------------------------------------------------------
__global__ void init_norm_kernel(float* __restrict__ deg, int n2) {
  int i = blockIdx.x * blockDim.x + threadIdx.x;
  if (i < n2) deg[i] = 1.0f;  // self-loop contributes 1 to every in/out degree
}

__global__ void count_deg_kernel(const int* __restrict__ src, const int* __restrict__ dst,
                                 float* __restrict__ outdeg, float* __restrict__ indeg, int E) {
  int e = blockIdx.x * blockDim.x + threadIdx.x;
  if (e < E) {
    atomicAdd(&outdeg[src[e]], 1.0f);
    atomicAdd(&indeg[dst[e]], 1.0f);
  }
}

__global__ void finish_norm_kernel(float* __restrict__ deg, int n2) {
  int i = blockIdx.x * blockDim.x + threadIdx.x;
  if (i < n2) deg[i] = rsqrtf(fmaxf(deg[i], 1.0f));
}

// ---------------------------------------------------------------------------
// One WMMA k-step: 16x16x4 f32, all T column tiles. B fragment is a single
// contiguous ds_load_b64 thanks to the paired LDS layout.
// ---------------------------------------------------------------------------
template <int COUT, int T>
__device__ __forceinline__ void wmma_step(const v2f* __restrict__ Wl, int pairIdx, int lrow,
                                          v2f a, v8f (&acc)[T]) {
#pragma unroll
  for (int t = 0; t < T; ++t) {
    v2f b = Wl[(size_t)pairIdx * COUT + t * 16 + lrow];
    acc[t] = __builtin_amdgcn_wmma_f32_16x16x4_f32(
        /*neg_a=*/false, a, /*neg_b=*/false, b,
        /*c_mod=*/(short)0, acc[t], /*reuse_a=*/false, /*reuse_b=*/false);
  }
}

// ---------------------------------------------------------------------------
// WMMA GEMM:  H[N,COUT] = (X * out_is[:,None]) @ W[128,COUT]
// f32 path: V_WMMA_F32_16X16X4_F32, exact f32 accumulate.
// One wave computes a 16-row stripe across all COUT columns (T=COUT/16 tiles),
// 8 waves per block. W is staged in LDS *pre-paired*: Wl[p][c] holds the
// (W[2p][c], W[2p+1][c]) v2f so every B fragment is one aligned ds_load_b64.
// A loads are software-pipelined (next k-step prefetched before the WMMAs).
// The result tile is stored to BOTH HA (scatter source) and HB (agg seeded
// with the self-loop term), fusing away the separate copy pass.
//
// Fragment layouts (ISA 7.12.2, wave32):
//   A 16x4 f32 : lanes 0-15 hold M=lane, {K=0,K=1}; lanes 16-31 hold {K=2,K=3}
//   B 4x16 f32 : lanes 0-15 hold N=lane, {K=0,K=1}; lanes 16-31 hold {K=2,K=3}
//   C/D 16x16  : 8 VGPRs, reg r: lanes 0-15 -> M=r, lanes 16-31 -> M=r+8
// ---------------------------------------------------------------------------
template <int COUT>
__global__ __launch_bounds__(256) void gemm_scaled_kernel(
    const float* __restrict__ X, const float* __restrict__ Wm,
    const float* __restrict__ out_is, float* __restrict__ HA,
    float* __restrict__ HB, int N) {
  constexpr int CIN = 128;
  constexpr int T = COUT / 16;
  constexpr int PAIRS = CIN / 2;  // 64 k-pairs
  __shared__ v2f Wl[PAIRS * COUT];

  // Stage W[CIN][COUT] into paired layout: Wl[p*COUT + c] = {W[2p][c], W[2p+1][c]}
  {
    float* Wf = (float*)Wl;
    for (int i = threadIdx.x; i < CIN * COUT; i += 256) {
      const int k = i / COUT;
      const int c = i - k * COUT;
      Wf[(((k >> 1) * COUT) + c) * 2 + (k & 1)] = Wm[i];
    }
  }
  __syncthreads();

  const int wave = threadIdx.x >> 5;
  const int lane = threadIdx.x & 31;
  const int half = lane >> 4;   // 0: K pair {0,1}, 1: K pair {2,3}
  const int lrow = lane & 15;

  const int tile = blockIdx.x * 8 + wave;
  const int row0 = tile * 16;
  if (row0 < N) {                       // wave-uniform guard (EXEC all-1 inside)
    const int arow = row0 + lrow;
    const float s = out_is[arow];
    // Per k-step kb, this lane needs X[arow][kb*4 + half*2 .. +1] == xp[kb*2+half]
    const v2f* __restrict__ xp = (const v2f*)(X + (size_t)arow * CIN);

    v8f acc[T] = {};

    v2f araw = xp[half];                // prefetch kb = 0
#pragma unroll 4
    for (int kb = 0; kb < CIN / 4 - 1; ++kb) {
      const v2f a = araw * s;
      araw = xp[(kb + 1) * 2 + half];   // prefetch next step before the WMMAs
      wmma_step<COUT, T>(Wl, kb * 2 + half, lrow, a, acc);
    }
    {  // peeled last step (no prefetch past end of row)
      const v2f a = araw * s;
      wmma_step<COUT, T>(Wl, (CIN / 4 - 1) * 2 + half, lrow, a, acc);
    }

    float* __restrict__ baseA = HA + (size_t)row0 * COUT;
    float* __restrict__ baseB = HB + (size_t)row0 * COUT;
#pragma unroll
    for (int t = 0; t < T; ++t) {
#pragma unroll
      for (int r = 0; r < 8; ++r) {
        const int m = half * 8 + r;
        const size_t off = (size_t)m * COUT + t * 16 + lrow;
        const float v = acc[t][r];
        baseA[off] = v;
        baseB[off] = v;   // agg seed == self-loop contribution
      }
    }
  }
}

// ---------------------------------------------------------------------------
// Edge scatter: one wave per edge; float4 gather of h[src], 4 no-return
// GLOBAL_ATOMIC_ADD_F32 per lane into agg[dst]. Both arrays are L2-resident
// (51 MB each vs 192 MB L2).
// ---------------------------------------------------------------------------
template <int C>
__global__ void scatter_kernel(const float* __restrict__ h, const int* __restrict__ src,
                               const int* __restrict__ dst, float* __restrict__ agg, int E) {
  const size_t tid = (size_t)blockIdx.x * blockDim.x + threadIdx.x;
  const int lane = threadIdx.x & 31;
  const size_t e = tid >> 5;
  if (e >= (size_t)E) return;
  const int s = src[e];
  const int d = dst[e];
  constexpr int NV = C / 4;
  if (lane < NV) {
    const float4 x = ((const float4*)(h + (size_t)s * C))[lane];
    float* ad = agg + (size_t)d * C + lane * 4;
    atomicAdd(ad + 0, x.x);
    atomicAdd(ad + 1, x.y);
    atomicAdd(ad + 2, x.z);
    atomicAdd(ad + 3, x.w);
  }
}

// ---------------------------------------------------------------------------
// Y = agg * in_is[:,None] + bias, optional ReLU (in place)
// ---------------------------------------------------------------------------
template <int C, bool RELU>
__global__ void finalize_kernel(float* __restrict__ Bm, const float* __restrict__ in_is,
                                const float* __restrict__ bias, int N) {
  constexpr int LOG = (C == 128) ? 7 : 6;
  const size_t i = (size_t)blockIdx.x * blockDim.x + threadIdx.x;
  const size_t total = (size_t)N * C;
  if (i >= total) return;
  const int f = (int)(i & (C - 1));
  const size_t row = i >> LOG;
  float v = Bm[i] * in_is[row] + bias[f];
  if (RELU) v = fmaxf(v, 0.0f);
  Bm[i] = v;
}

// ---------------------------------------------------------------------------
// Column mean over N rows of a [N,64] matrix -> out[64]
// ---------------------------------------------------------------------------
__global__ void zero_out_kernel(float* __restrict__ out) { out[threadIdx.x] = 0.0f; }

__global__ void mean_kernel(const float* __restrict__ Y, float* __restrict__ out,
                            int N, float invN) {
  __shared__ float sd[256];
  const int f = threadIdx.x & 63;
  const int g = threadIdx.x >> 6;  // 0..3 row groups per block
  float acc = 0.0f;
  for (int i = blockIdx.x * 4 + g; i < N; i += gridDim.x * 4)
    acc += Y[(size_t)i * 64 + f];
  sd[threadIdx.x] = acc;
  __syncthreads();
  if (threadIdx.x < 64) {
    const float tot = sd[threadIdx.x] + sd[threadIdx.x + 64] +
                      sd[threadIdx.x + 128] + sd[threadIdx.x + 192];
    atomicAdd(&out[threadIdx.x], tot * invN);
  }
}

// ---------------------------------------------------------------------------
// Host-side orchestration
// ---------------------------------------------------------------------------
extern "C" void kernel_launch(void* const* d_in, const int* in_sizes, int n_in,
                              void* d_out, int out_size, void* d_ws, size_t ws_size,
                              hipStream_t stream) {
  const float* feat = (const float*)d_in[0];
  const int* src = (const int*)d_in[1];
  const int* dst = (const int*)d_in[2];
  const float* W1 = (const float*)d_in[3];
  const float* b1 = (const float*)d_in[4];
  const float* W2 = (const float*)d_in[5];
  const float* b2 = (const float*)d_in[6];
  const float* W3 = (const float*)d_in[7];
  const float* b3 = (const float*)d_in[8];

  const int FIN = 128;
  const int N = in_sizes[0] / FIN;   // 100000
  const int E = in_sizes[1];         // 1600000
  float* out = (float*)d_out;        // [64]

  float* ws = (float*)d_ws;
  float* out_is = ws;                         // [N]
  float* in_is = ws + N;                      // [N]
  float* A = ws + 2 * (size_t)N;              // [N,128] scratch (h = Xs @ W)
  float* B = A + (size_t)N * 128;             // [N,128] scratch (agg / output)

  // --- degrees & symmetric normalization ---
  init_norm_kernel<<<(2 * N + 255) / 256, 256, 0, stream>>>(out_is, 2 * N);
  count_deg_kernel<<<(E + 255) / 256, 256, 0, stream>>>(src, dst, out_is, in_is, E);
  finish_norm_kernel<<<(2 * N + 255) / 256, 256, 0, stream>>>(out_is, 2 * N);

  const int gemmBlocks = ((N + 15) / 16 + 7) / 8;
  const int scatBlocks = (int)(((size_t)E * 32 + 255) / 256);

  // --- layer 1: feat -> B (relu) ---
  gemm_scaled_kernel<128><<<gemmBlocks, 256, 0, stream>>>(feat, W1, out_is, A, B, N);
  scatter_kernel<128><<<scatBlocks, 256, 0, stream>>>(A, src, dst, B, E);
  finalize_kernel<128, true><<<((size_t)N * 128 + 255) / 256, 256, 0, stream>>>(B, in_is, b1, N);

  // --- layer 2: B -> B (relu) ---
  gemm_scaled_kernel<128><<<gemmBlocks, 256, 0, stream>>>(B, W2, out_is, A, B, N);
  scatter_kernel<128><<<scatBlocks, 256, 0, stream>>>(A, src, dst, B, E);
  finalize_kernel<128, true><<<((size_t)N * 128 + 255) / 256, 256, 0, stream>>>(B, in_is, b2, N);

  // --- layer 3: B -> B[N,64] (no relu) ---
  gemm_scaled_kernel<64><<<gemmBlocks, 256, 0, stream>>>(B, W3, out_is, A, B, N);
  scatter_kernel<64><<<scatBlocks, 256, 0, stream>>>(A, src, dst, B, E);
  finalize_kernel<64, false><<<((size_t)N * 64 + 255) / 256, 256, 0, stream>>>(B, in_is, b3, N);

  // --- mean over nodes ---
  zero_out_kernel<<<1, 64, 0, stream>>>(out);
  mean_kernel<<<256, 256, 0, stream>>>(B, out, N, 1.0f / (float)N);
}